// mDACF_72808285602364
// MI455X (gfx1250) — compile-verified
//
#include <hip/hip_runtime.h>

typedef float v2f __attribute__((ext_vector_type(2)));
typedef float v4f __attribute__((ext_vector_type(4)));
typedef float v8f __attribute__((ext_vector_type(8)));

#define FDIM 512              // feature dim (K)
#define LDIM 256              // latent dim (N)
#define PITCH 516             // LDS row pitch in floats: 516 % 64 == 4 -> conflict-free A reads
#define MT 32                 // M-tile rows per block (2 sub-tiles of 16)

extern __shared__ float lds[];

__global__ __launch_bounds__(256)
void two_tower_fused_kernel(const int* __restrict__ x,
                            const float* __restrict__ user_lookup,
                            const float* __restrict__ item_lookup,
                            const float* __restrict__ user_W,
                            const float* __restrict__ user_b,
                            const float* __restrict__ item_W,
                            const float* __restrict__ item_b,
                            float* __restrict__ out)
{
    const int tid = threadIdx.x;
    const int m0  = blockIdx.x * MT;        // 32 batch rows per block

    // ---- Stage 32 user rows + 32 item rows into LDS (random gather) ----
    {
        const int row = tid >> 2;           // 0..63 (0..31 user, 32..63 item)
        const int l4  = tid & 3;            // 4 threads cooperate per row
        const int idx = (row < MT) ? x[2 * (m0 + row)]
                                   : x[2 * (m0 + row - MT) + 1];
        const float* src = (row < MT ? user_lookup : item_lookup) + (size_t)idx * FDIM;
        float* dst = lds + row * PITCH;
        #pragma unroll
        for (int j = 0; j < 32; ++j) {
            const int w = (j * 4 + l4) * 4;           // float4 granularity, coalesced
            *(v4f*)(dst + w) = *(const v4f*)(src + w);
        }
    }
    __syncthreads();

    const int lane = tid & 31;
    const int wave = tid >> 5;
    const int l16  = lane & 15;
    const int hi   = lane >> 4;             // 0: K={k,k+1}, 1: K={k+2,k+3} (A/B frag layout)

    // each wave owns latent columns [32*wave, 32*wave+32) for BOTH towers
    const int col0 = wave * 32 + l16;
    const int col1 = col0 + 16;

    // A fragment bases: sub-tile s covers rows 16s..16s+15 of the M-tile
    const float* uA0 = lds + (      l16) * PITCH + 2 * hi;
    const float* uA1 = lds + (16  + l16) * PITCH + 2 * hi;
    const float* iA0 = lds + (MT      + l16) * PITCH + 2 * hi;
    const float* iA1 = lds + (MT + 16 + l16) * PITCH + 2 * hi;

    const float* uB0 = user_W + (size_t)col0 * FDIM + 2 * hi;   // weight B fragments (global/L2)
    const float* uB1 = user_W + (size_t)col1 * FDIM + 2 * hi;
    const float* iB0 = item_W + (size_t)col0 * FDIM + 2 * hi;
    const float* iB1 = item_W + (size_t)col1 * FDIM + 2 * hi;

    v8f cu00 = {}, cu01 = {}, cu10 = {}, cu11 = {};
    v8f ci00 = {}, ci01 = {}, ci10 = {}, ci11 = {};

    #pragma unroll 4
    for (int k = 0; k < FDIM; k += 4) {
        v2f au0 = *(const v2f*)(uA0 + k);
        v2f au1 = *(const v2f*)(uA1 + k);
        v2f ai0 = *(const v2f*)(iA0 + k);
        v2f ai1 = *(const v2f*)(iA1 + k);
        v2f bu0 = *(const v2f*)(uB0 + k);
        v2f bu1 = *(const v2f*)(uB1 + k);
        v2f bi0 = *(const v2f*)(iB0 + k);
        v2f bi1 = *(const v2f*)(iB1 + k);
        cu00 = __builtin_amdgcn_wmma_f32_16x16x4_f32(false, au0, false, bu0, (short)0, cu00, false, false);
        cu01 = __builtin_amdgcn_wmma_f32_16x16x4_f32(false, au0, false, bu1, (short)0, cu01, false, false);
        cu10 = __builtin_amdgcn_wmma_f32_16x16x4_f32(false, au1, false, bu0, (short)0, cu10, false, false);
        cu11 = __builtin_amdgcn_wmma_f32_16x16x4_f32(false, au1, false, bu1, (short)0, cu11, false, false);
        ci00 = __builtin_amdgcn_wmma_f32_16x16x4_f32(false, ai0, false, bi0, (short)0, ci00, false, false);
        ci01 = __builtin_amdgcn_wmma_f32_16x16x4_f32(false, ai0, false, bi1, (short)0, ci01, false, false);
        ci10 = __builtin_amdgcn_wmma_f32_16x16x4_f32(false, ai1, false, bi0, (short)0, ci10, false, false);
        ci11 = __builtin_amdgcn_wmma_f32_16x16x4_f32(false, ai1, false, bi1, (short)0, ci11, false, false);
    }

    // ---- bias + relu + elementwise product (fused dot partials, per sub-tile) ----
    const float bu0s = user_b[col0], bu1s = user_b[col1];
    const float bi0s = item_b[col0], bi1s = item_b[col1];

    v8f acc0, acc1;
    #pragma unroll
    for (int e = 0; e < 8; ++e) {
        float a0 = fmaxf(cu00[e] + bu0s, 0.0f) * fmaxf(ci00[e] + bi0s, 0.0f);
        float a1 = fmaxf(cu01[e] + bu1s, 0.0f) * fmaxf(ci01[e] + bi1s, 0.0f);
        acc0[e] = a0 + a1;
        float b0 = fmaxf(cu10[e] + bu0s, 0.0f) * fmaxf(ci10[e] + bi0s, 0.0f);
        float b1 = fmaxf(cu11[e] + bu1s, 0.0f) * fmaxf(ci11[e] + bi1s, 0.0f);
        acc1[e] = b0 + b1;
    }

    // reduce over the 16 columns held by each 16-lane group (halves stay disjoint)
    #pragma unroll
    for (int m = 1; m < 16; m <<= 1) {
        #pragma unroll
        for (int e = 0; e < 8; ++e) {
            acc0[e] += __shfl_xor(acc0[e], m, 32);
            acc1[e] += __shfl_xor(acc1[e], m, 32);
        }
    }
    // lane 0 holds rows {e} (sub-tile 0) / {16+e} (sub-tile 1); lane 16 holds rows {8+e} / {24+e}

    // ---- cross-wave reduction via LDS (reuse staging memory) ----
    __syncthreads();
    float* red = lds;
    if (tid < MT) red[tid] = 0.0f;
    __syncthreads();
    if (lane == 0) {
        #pragma unroll
        for (int e = 0; e < 8; ++e) {
            atomicAdd(&red[e],      acc0[e]);
            atomicAdd(&red[16 + e], acc1[e]);
        }
    } else if (lane == 16) {
        #pragma unroll
        for (int e = 0; e < 8; ++e) {
            atomicAdd(&red[8 + e],  acc0[e]);
            atomicAdd(&red[24 + e], acc1[e]);
        }
    }
    __syncthreads();
    if (tid < MT) out[m0 + tid] = red[tid];
}

extern "C" void kernel_launch(void* const* d_in, const int* in_sizes, int n_in,
                              void* d_out, int out_size, void* d_ws, size_t ws_size,
                              hipStream_t stream) {
    const int*   x           = (const int*)d_in[0];
    const float* user_lookup = (const float*)d_in[1];
    const float* item_lookup = (const float*)d_in[2];
    const float* user_W      = (const float*)d_in[3];
    const float* user_b      = (const float*)d_in[4];
    const float* item_W      = (const float*)d_in[5];
    const float* item_b      = (const float*)d_in[6];
    float* out = (float*)d_out;

    const int B = in_sizes[0] / 2;                            // 16384
    const size_t shmem = (size_t)(2 * MT) * PITCH * sizeof(float); // 132096 B dynamic LDS

    // allow >64KB dynamic LDS (WGP has 320KB); deterministic, safe under graph capture
    (void)hipFuncSetAttribute((const void*)two_tower_fused_kernel,
                              hipFuncAttributeMaxDynamicSharedMemorySize, (int)shmem);

    dim3 grid(B / MT);
    dim3 block(256);
    two_tower_fused_kernel<<<grid, block, shmem, stream>>>(
        x, user_lookup, item_lookup, user_W, user_b, item_W, item_b, out);
}